// ProposalTarget_5531917877593
// MI455X (gfx1250) — compile-verified
//
#include <hip/hip_runtime.h>
#include <stdint.h>

// Faster-RCNN ProposalTarget for MI455X (gfx1250, wave32).
// Phase 1: IoU + argmax; gt table staged to LDS via CDNA5 async global->LDS
//          DMA (ASYNCcnt path), division-free inner loop, float4 LDS broadcast.
// Phase 2: exact stable top-32/top-96 via wave32 shuffle reductions (2 barriers
//          per selection round), then gather/transform/scatter outputs.

#define NUM_CLASSES 21
#define N_SAMPLE    128
#define N_POS       32
#define N_NEG       96
#define KMAX        128
#define T1          256
#define T2          1024

typedef unsigned long long u64;

// Monotonic float->uint mapping: a > b  <=>  map(a) > map(b)
__device__ __forceinline__ unsigned key_map(float f) {
  unsigned b = __float_as_uint(f);
  return (b & 0x80000000u) ? ~b : (b | 0x80000000u);
}
// Pack (key, index) so u64 max == stable top-k (lower index wins ties,
// matching jax.lax.top_k).
__device__ __forceinline__ u64 pack_key(float key, unsigned idx) {
  return ((u64)key_map(key) << 32) | (u64)(0xFFFFFFFFu - idx);
}

// Byte offset of a __shared__ object inside LDS (async-to-LDS dst VGPR).
__device__ __forceinline__ unsigned lds_offset_of(const void* p) {
  return (unsigned)(uintptr_t)(const __attribute__((address_space(3))) char*)p;
}

__global__ void __launch_bounds__(T1)
iou_argmax_kernel(const float* __restrict__ props, const float* __restrict__ gtb,
                  const float* __restrict__ u, u64* __restrict__ posP,
                  u64* __restrict__ negP, int* __restrict__ amax,
                  int N, int K, int NB) {
  __shared__ float  graw[KMAX * 5];
  __shared__ float4 gbox[KMAX];   // {x0, y0, x1+1, y1+1} -> one ds_load_b128
  __shared__ float  sarea[KMAX];
  const int tid = threadIdx.x;

  // --- CDNA5 async DMA: stage gt table (K*5 floats) into LDS, 16B per lane ---
  const int nvec = (K * 5 + 3) >> 2;  // b128 chunks (=160 for K=128)
  if (tid < nvec) {
    unsigned ldsoff = lds_offset_of(&graw[tid * 4]);
    u64 gaddr = (u64)(uintptr_t)(gtb + tid * 4);
    asm volatile("global_load_async_to_lds_b128 %0, %1, off"
                 :: "v"(ldsoff), "v"(gaddr) : "memory");
  }
  asm volatile("s_wait_asynccnt 0x0" ::: "memory");
  __syncthreads();

  if (tid < K) {  // repack: +1.0 folded into far corner; precomputed area
    float x0 = graw[tid * 5 + 0], y0 = graw[tid * 5 + 1];
    float x1 = graw[tid * 5 + 2], y1 = graw[tid * 5 + 3];
    gbox[tid] = make_float4(x0, y0, x1 + 1.f, y1 + 1.f);
    sarea[tid] = (x1 - x0 + 1.f) * (y1 - y0 + 1.f);
  }
  __syncthreads();

  const int i = blockIdx.x * T1 + tid;
  if (i >= NB) return;

  float bx0, by0, bx1, by1;
  if (i < N) {
    float4 b = ((const float4*)props)[i];
    bx0 = b.x; by0 = b.y; bx1 = b.z; by1 = b.w;
  } else {
    const float* g = gtb + (u64)(i - N) * 5;
    bx0 = g[0]; by0 = g[1]; bx1 = g[2]; by1 = g[3];
  }
  const float areaB = (bx1 - bx0 + 1.f) * (by1 - by0 + 1.f);
  const float bx1p = bx1 + 1.f, by1p = by1 + 1.f;

  // Division-free running argmax: iou_g > iou_best  <=>
  //   inter_g * denom_best > inter_best * denom_g   (denoms > 0).
  float bInter = -1.f, bDenom = 1.f;
  int bIdx = 0;
#pragma unroll 4
  for (int g = 0; g < K; ++g) {
    float4 gb = gbox[g];                 // LDS broadcast, conflict-free
    float aSum = areaB + sarea[g];
    float w = fminf(bx1p, gb.z) - fmaxf(bx0, gb.x);  // = min(x1)+1 - max(x0)
    float h = fminf(by1p, gb.w) - fmaxf(by0, gb.y);
    w = fmaxf(w, 0.f); h = fmaxf(h, 0.f);
    float inter = w * h;
    float denom = aSum - inter;
    if (inter * bDenom > bInter * denom) { bInter = inter; bDenom = denom; bIdx = g; }
  }
  const float maxov = bInter / bDenom;  // single division per box
  const float uu = u[i];
  const float pk = (maxov > 0.5f) ? uu : -1.f;
  const float nk = (maxov < 0.3f && maxov > 0.1f) ? uu : -1.f;
  posP[i] = pack_key(pk, (unsigned)i);
  negP[i] = pack_key(nk, (unsigned)i);
  amax[i] = bIdx;
}

// Top-4 of this thread's strided chunk among elements strictly below `thresh`.
__device__ __forceinline__ void scan_top4(const u64* __restrict__ a, int tid,
                                          int NB, u64 thresh,
                                          u64& c0, u64& c1, u64& c2, u64& c3) {
  c0 = c1 = c2 = c3 = 0ull;
  for (int i = tid; i < NB; i += T2) {  // fully coalesced per iteration
    u64 v = a[i];
    if (v < thresh && v > c3) {
      if (v > c0)      { c3 = c2; c2 = c1; c1 = c0; c0 = v; }
      else if (v > c1) { c3 = c2; c2 = c1; c1 = v; }
      else if (v > c2) { c3 = c2; c2 = v; }
      else             { c3 = v; }
    }
  }
}

__global__ void __launch_bounds__(T2)
select_finalize_kernel(const float* __restrict__ props, const float* __restrict__ gtb,
                       const u64* __restrict__ posP, const u64* __restrict__ negP,
                       const int* __restrict__ amax, float* __restrict__ out,
                       int N, int NB) {
  __shared__ u64 wvals[T2 / 32];
  __shared__ int wown[T2 / 32];
  __shared__ int winO;
  __shared__ unsigned selS[N_SAMPLE];
  const int tid = threadIdx.x;
  const int lane = tid & 31, wave = tid >> 5;

  u64 c0, c1, c2, c3;
  const u64* arr = posP;
  scan_top4(arr, tid, NB, ~0ull, c0, c1, c2, c3);

  for (int r = 0; r < N_SAMPLE; ++r) {
    if (r == N_POS) {  // switch to negative keys
      arr = negP;
      scan_top4(arr, tid, NB, ~0ull, c0, c1, c2, c3);
    }
    // Stage 1: wave32 shuffle max-reduce of (candidate, owner-tid).
    u64 v = c0; int o = tid;
#pragma unroll
    for (int off = 16; off > 0; off >>= 1) {
      u64 v2 = __shfl_down(v, off, 32);
      int o2 = __shfl_down(o, off, 32);
      if (v2 > v) { v = v2; o = o2; }
    }
    if (lane == 0) { wvals[wave] = v; wown[wave] = o; }
    __syncthreads();
    // Stage 2: wave 0 reduces the 32 wave winners.
    if (wave == 0) {
      u64 wv = wvals[lane]; int wo = wown[lane];
#pragma unroll
      for (int off = 16; off > 0; off >>= 1) {
        u64 v2 = __shfl_down(wv, off, 32);
        int o2 = __shfl_down(wo, off, 32);
        if (v2 > wv) { wv = v2; wo = o2; }
      }
      if (lane == 0) {
        winO = wo;
        selS[r] = 0xFFFFFFFFu - (unsigned)(wv & 0xFFFFFFFFull);
      }
    }
    __syncthreads();  // winO/selS visible; also orders wvals reuse next round
    if (tid == winO) {  // pop winner; refill from L2 only if cache exhausted
      u64 th = c0;
      c0 = c1; c1 = c2; c2 = c3; c3 = 0ull;
      if (c0 == 0ull) scan_top4(arr, tid, NB, th, c0, c1, c2, c3);
    }
  }

  // Zero-init bbox_target region (harness poisons d_out).
  for (int j = tid; j < N_SAMPLE * NUM_CLASSES * 4; j += T2)
    out[N_SAMPLE * 4 + N_SAMPLE + j] = 0.f;
  __syncthreads();

  if (tid < N_SAMPLE) {
    int i = (int)selS[tid];
    float bx0, by0, bx1, by1;
    if (i < N) {
      const float* p = props + (u64)i * 4;
      bx0 = p[0]; by0 = p[1]; bx1 = p[2]; by1 = p[3];
    } else {
      const float* g = gtb + (u64)(i - N) * 5;
      bx0 = g[0]; by0 = g[1]; bx1 = g[2]; by1 = g[3];
    }
    out[tid * 4 + 0] = bx0; out[tid * 4 + 1] = by0;
    out[tid * 4 + 2] = bx1; out[tid * 4 + 3] = by1;

    int am = amax[i];
    int label = (tid < N_POS) ? (int)gtb[(u64)am * 5 + 4] : 0;
    out[N_SAMPLE * 4 + tid] = (float)label;

    if (label >= 1) {  // bbox_transform + scatter into class slot
      const float* g = gtb + (u64)am * 5;
      float ew = bx1 - bx0 + 1.f, eh = by1 - by0 + 1.f;
      float ecx = bx0 + 0.5f * ew, ecy = by0 + 0.5f * eh;
      float gw = g[2] - g[0] + 1.f, gh = g[3] - g[1] + 1.f;
      float gcx = g[0] + 0.5f * gw, gcy = g[1] + 0.5f * gh;
      int base = N_SAMPLE * 4 + N_SAMPLE + tid * (NUM_CLASSES * 4) + label * 4;
      out[base + 0] = (gcx - ecx) / ew;
      out[base + 1] = (gcy - ecy) / eh;
      out[base + 2] = logf(gw / ew);
      out[base + 3] = logf(gh / eh);
    }
  }
}

extern "C" void kernel_launch(void* const* d_in, const int* in_sizes, int n_in,
                              void* d_out, int out_size, void* d_ws, size_t ws_size,
                              hipStream_t stream) {
  const float* props = (const float*)d_in[0];  // (1, N, 4) f32
  const float* gtb   = (const float*)d_in[1];  // (1, K, 5) f32
  const float* u     = (const float*)d_in[2];  // (N+K,)   f32
  const int N  = in_sizes[0] / 4;
  const int K  = in_sizes[1] / 5;
  const int NB = in_sizes[2];

  // Workspace: pos keys (NB u64) | neg keys (NB u64) | argmax (NB i32) ~= 4 MB
  u64* posP = (u64*)d_ws;
  u64* negP = posP + NB;
  int* amax = (int*)(negP + NB);

  const int blocks = (NB + T1 - 1) / T1;
  iou_argmax_kernel<<<blocks, T1, 0, stream>>>(props, gtb, u, posP, negP, amax,
                                               N, K, NB);
  select_finalize_kernel<<<1, T2, 0, stream>>>(props, gtb, posP, negP, amax,
                                               (float*)d_out, N, NB);
}